// OurPolicy_71193377898773
// MI455X (gfx1250) — compile-verified
//
#include <hip/hip_runtime.h>

// Problem: out = copy(aw[1,16,2048,2048]) with last row of each head possibly
// replaced per argmax-histogram logic. Pure HBM-bound copy (~512MB traffic,
// ~22us floor at 23.3 TB/s) + trivial scalar decision work.

#define HEADS   16
#define SEQ     2048
#define KCONST  4

// ---------------------------------------------------------------------------
// Async global<->LDS copy path (CDNA5 ASYNCcnt engine), with safe fallback.
// Probe-verified: builtins exist; param 1 is AS(1) int-vector-16B pointer.
// ---------------------------------------------------------------------------
#if __has_builtin(__builtin_amdgcn_global_load_async_to_lds_b128) && \
    __has_builtin(__builtin_amdgcn_global_store_async_from_lds_b128)
#define USE_ASYNC_LDS 1
#else
#define USE_ASYNC_LDS 0
#endif

typedef int v4i __attribute__((vector_size(16)));
typedef __attribute__((address_space(1))) v4i* as1_v4i_ptr;
typedef __attribute__((address_space(3))) v4i* as3_v4i_ptr;

static __device__ __forceinline__ void wait_asynccnt0() {
#if __has_builtin(__builtin_amdgcn_s_wait_asynccnt)
  __builtin_amdgcn_s_wait_asynccnt(0);
#else
  asm volatile("s_wait_asynccnt 0" ::: "memory");
#endif
}

// -------------------------- Kernel 1: per-head argmax ----------------------
// One block per head. Each thread scans a contiguous chunk of 8 elements
// (first-max within chunk), then LDS tree-reduce with lower-index tie-break
// => exact jnp.argmax (first occurrence) semantics.
__global__ void argmax_kernel(const float* __restrict__ aw, int* __restrict__ ws) {
  const int h = blockIdx.x;
  const float* __restrict__ row = aw + ((size_t)h * SEQ + (SEQ - 1)) * SEQ;

  __shared__ float sval[256];
  __shared__ int   sidx[256];

  const int t = threadIdx.x;
  const int base = t * (SEQ / 256);           // 8 contiguous elements per thread
  float best = -3.402823466e38f;
  int   bidx = base;
#pragma unroll
  for (int j = 0; j < SEQ / 256; ++j) {
    float v = row[base + j];
    if (v > best) { best = v; bidx = base + j; }
  }
  sval[t] = best;
  sidx[t] = bidx;
  __syncthreads();

  for (int s = 128; s > 0; s >>= 1) {
    if (t < s) {
      float v2 = sval[t + s];
      int   i2 = sidx[t + s];
      if (v2 > sval[t] || (v2 == sval[t] && i2 < sidx[t])) {
        sval[t] = v2;
        sidx[t] = i2;
      }
    }
    __syncthreads();
  }
  if (t == 0) ws[h] = sidx[0];
}

// -------------------------- Kernel 2: merge decision -----------------------
// ws[0..15] = cand  ->  ws[16..31] = source head for each head's last row.
__global__ void decide_kernel(int* __restrict__ ws) {
  if (threadIdx.x != 0) return;
  int cand[HEADS];
#pragma unroll
  for (int h = 0; h < HEADS; ++h) cand[h] = ws[h];

  int cnt[HEADS];
  int maxv = 0;
#pragma unroll
  for (int h = 0; h < HEADS; ++h) {
    int c = 0;
#pragma unroll
    for (int g = 0; g < HEADS; ++g) c += (cand[g] == cand[h]) ? 1 : 0;
    cnt[h] = c;
    maxv = (c > maxv) ? c : maxv;
  }
  // sampled = first head whose count equals the max (argmax over bool mask)
  int sampled = 0;
  for (int h = HEADS - 1; h >= 0; --h)
    if (cnt[h] == maxv) sampled = h;

  const bool do_merge = (maxv <= KCONST);
#pragma unroll
  for (int h = 0; h < HEADS; ++h) {
    const bool mask = (cnt[h] == maxv);
    ws[HEADS + h] = (mask && do_merge) ? sampled : h;
  }
}

// -------------------------- Kernel 3: bulk copy ----------------------------
// 64 bytes per thread (4 x b128). Async path streams global->LDS->global via
// the ASYNCcnt engine without touching VGPRs; fallback is plain float4 copy.
#define COPY_ITEMS 4
__global__ void copy_kernel(const float4* __restrict__ src, float4* __restrict__ dst) {
  const size_t base = (size_t)blockIdx.x * (256 * COPY_ITEMS) + threadIdx.x;
#if USE_ASYNC_LDS
  __shared__ float4 buf[256 * COPY_ITEMS];
#pragma unroll
  for (int k = 0; k < COPY_ITEMS; ++k) {
    __builtin_amdgcn_global_load_async_to_lds_b128(
        (as1_v4i_ptr)(src + base + (size_t)k * 256),
        (as3_v4i_ptr)(&buf[threadIdx.x + k * 256]),
        0, 0);
  }
  wait_asynccnt0();
#pragma unroll
  for (int k = 0; k < COPY_ITEMS; ++k) {
    __builtin_amdgcn_global_store_async_from_lds_b128(
        (as1_v4i_ptr)(dst + base + (size_t)k * 256),
        (as3_v4i_ptr)(&buf[threadIdx.x + k * 256]),
        0, 0);
  }
  // s_endpgm performs an implicit wait-idle; stores drain before the next
  // stream-ordered kernel (fixup) can observe d_out.
#else
#pragma unroll
  for (int k = 0; k < COPY_ITEMS; ++k) {
    dst[base + (size_t)k * 256] = src[base + (size_t)k * 256];
  }
#endif
}

// -------------------------- Kernel 4: last-row fixup -----------------------
// Overwrite each head's last row in the output with the chosen source head's
// last row from the (unmodified) input. 128KB total; negligible.
__global__ void fixup_kernel(const float* __restrict__ aw, float* __restrict__ out,
                             const int* __restrict__ ws) {
  const int h   = blockIdx.x;
  const int src = ws[HEADS + h];
  const float4* __restrict__ srow =
      (const float4*)(aw + ((size_t)src * SEQ + (SEQ - 1)) * SEQ);
  float4* __restrict__ drow =
      (float4*)(out + ((size_t)h * SEQ + (SEQ - 1)) * SEQ);
  const int t = threadIdx.x;        // 256 threads, 512 float4s per row
  drow[t]       = srow[t];
  drow[t + 256] = srow[t + 256];
}

// ---------------------------------------------------------------------------
extern "C" void kernel_launch(void* const* d_in, const int* in_sizes, int n_in,
                              void* d_out, int out_size, void* d_ws, size_t ws_size,
                              hipStream_t stream) {
  (void)in_sizes; (void)n_in; (void)out_size; (void)ws_size;

  const float* aw  = (const float*)d_in[0];
  float*       out = (float*)d_out;
  int*         ws  = (int*)d_ws;

  // 1) per-head argmax of last rows -> ws[0..15]
  argmax_kernel<<<HEADS, 256, 0, stream>>>(aw, ws);
  // 2) histogram / mask / sampled-head decision -> ws[16..31]
  decide_kernel<<<1, 32, 0, stream>>>(ws);
  // 3) bulk copy of the whole tensor: 16*2048*2048 floats = 16Mi float4
  //    16Mi / (256 threads * 4 items) = 16384 blocks, exact (no tail)
  copy_kernel<<<16384, 256, 0, stream>>>((const float4*)aw, (float4*)out);
  // 4) rewrite the 16 last rows from their selected source heads
  fixup_kernel<<<HEADS, 256, 0, stream>>>(aw, out, ws);
}